// NucleusFieldSampler_63651415327275
// MI455X (gfx1250) — compile-verified
//
#include <hip/hip_runtime.h>
#include <hip/hip_bf16.h>

#define NBINS 8192
#define BLOCK 256
#define SEGSZ (NBINS / BLOCK)   // 32 bins per thread segment
#define TOPP  0.9f

typedef unsigned int v4u __attribute__((ext_vector_type(4)));
typedef int          v8i __attribute__((ext_vector_type(8)));
typedef int          v4i __attribute__((ext_vector_type(4)));

#if defined(__has_builtin)
#if __has_builtin(__builtin_amdgcn_tensor_load_to_lds)
#define HAVE_TDM 1
#endif
#endif

// Deterministic counter-based Gumbel noise (splitmix64 -> uniform -> gumbel).
__device__ __forceinline__ float gumbel_at(unsigned long long idx) {
  unsigned long long x = idx * 0x9E3779B97F4A7C15ull + 0xD1B54A32D192ED03ull;
  x = (x ^ (x >> 30)) * 0xBF58476D1CE4E5B9ull;
  x = (x ^ (x >> 27)) * 0x94D049BB133111EBull;
  x ^= (x >> 31);
  float u = ((float)(unsigned)(x >> 40) + 0.5f) * (1.0f / 16777216.0f);
  return -__logf(-__logf(u));
}

#ifdef HAVE_TDM
// Issue one 1-D TDM tile load (nelem f32, nelem <= 65535) global -> LDS.
__device__ __forceinline__ void tdm_issue_1d(const float* src, float* dst, int nelem) {
  unsigned long long ga = (unsigned long long)(uintptr_t)src;
  unsigned lds_off = (unsigned)(uintptr_t)(void*)dst;   // low 32 bits = LDS byte offset
  v4u g0;                                                // D# group 0 (ISA 8.3)
  g0.x = 1u;                                             // count=1, user mode
  g0.y = lds_off;                                        // lds_addr
  g0.z = (unsigned)(ga & 0xFFFFFFFFull);                 // global_addr[31:0]
  g0.w = (unsigned)((ga >> 32) & 0x01FFFFFFull) | 0x80000000u;  // ga[56:32] | type=2
  v8i g1;                                                // D# group 1 (ISA 8.4)
  g1[0] = (int)(2u << 16);                               // mask=0, data_size=2 (4B)
  g1[1] = (int)(((unsigned)nelem & 0xFFFFu) << 16);      // tensor_dim0 lo16
  g1[2] = (int)((((unsigned)nelem >> 16) & 0xFFFFu) | (1u << 16)); // td0 hi | td1=1
  g1[3] = (int)(((unsigned)nelem & 0xFFFFu) << 16);      // tile_dim0 = nelem
  g1[4] = 0;                                             // tile_dim1=0 (1-D), tile_dim2=0
  g1[5] = nelem;                                         // tensor_dim0_stride lo32
  g1[6] = 0;
  g1[7] = 0;
  v4i g2 = {0, 0, 0, 0};
  v4i g3 = {0, 0, 0, 0};
  v8i g4 = {0, 0, 0, 0, 0, 0, 0, 0};
  __builtin_amdgcn_tensor_load_to_lds(g0, g1, g2, g3, g4, 0);
}
#endif

// Parallel top-down cut finder: find bin b* where the suffix-cumsum of hist
// first exceeds budget, and the residual mass R available inside bin b*.
// All BLOCK threads participate; result in *s_bstar / *s_R.
__device__ __forceinline__ void find_cut(const float* hist, float budget,
                                         float* s_suf, int* s_bstar, float* s_R,
                                         int tid) {
  float seg = 0.0f;
  const int base = tid * SEGSZ;
  for (int i = 0; i < SEGSZ; ++i) seg += hist[base + i];
  s_suf[tid] = seg;
  __syncthreads();
  for (int off = 1; off < BLOCK; off <<= 1) {      // 8-step suffix-sum doubling
    float v = (tid + off < BLOCK) ? s_suf[tid + off] : 0.0f;
    __syncthreads();
    s_suf[tid] += v;
    __syncthreads();
  }
  if (tid == 0) { *s_bstar = 0; *s_R = 1e30f; }    // default: everything in nucleus
  __syncthreads();
  const float suf = s_suf[tid];                    // mass in bins >= base
  const float above = suf - seg;                   // mass in segments above this one
  if (suf > budget && above <= budget) {           // crossing is inside my segment
    float acc = above;
    int bsel = base; float Rsel = budget - acc;
    for (int b = base + SEGSZ - 1; b >= base; --b) {
      float h = hist[b];
      if (acc + h > budget) { bsel = b; Rsel = budget - acc; break; }
      acc += h;
    }
    *s_bstar = bsel; *s_R = Rsel;
  }
  __syncthreads();
}

__global__ void __launch_bounds__(BLOCK)
nucleus_sample_kernel(const float* __restrict__ probs, int V, int* __restrict__ out)
{
  extern __shared__ float smem[];
  float* rowbuf = smem;        // V floats: whole probability row, DMA'd by TDM
  float* hist   = smem + V;    // NBINS float bins (reused coarse then refined)

  __shared__ float    s_suf[BLOCK];
  __shared__ float    s_sum;
  __shared__ unsigned s_maxbits;
  __shared__ int      s_bstar;
  __shared__ float    s_R;
  __shared__ float    s_wbest[BLOCK / 32];
  __shared__ int      s_wbidx[BLOCK / 32];

  const int tid   = threadIdx.x;
  const int rowId = blockIdx.x;
  const float* __restrict__ src = probs + (size_t)rowId * (size_t)V;

  for (int b = tid; b < NBINS; b += BLOCK) hist[b] = 0.0f;
  if (tid == 0) { s_sum = 0.0f; s_maxbits = 0u; }

  // Pass-1 worker: sum + coarse histogram (bits>>17) + row max, over [lo,hi).
  float lsum = 0.0f;
  unsigned lmax = 0u;
  auto pass1 = [&](int lo, int hi) {
    for (int v = lo + tid; v < hi; v += BLOCK) {
      float p = rowbuf[v];
      unsigned bits = __float_as_uint(p);
      lsum += p;
      if (bits > lmax) lmax = bits;
      atomicAdd(&hist[bits >> 17], p);     // ds_add_f32; p<1 => bin < 8128
    }
  };

  // Chunk boundaries for 4-way DMA/compute pipelining.
  const int chunk = (V + 3) >> 2;
  const int b1 = min(chunk, V);
  const int b2 = min(2 * chunk, V);
  const int b3 = min(3 * chunk, V);

#ifdef HAVE_TDM
  // ---- TDM: 4 chunked async DMA loads of the row (201 KB total) ----
  if (tid < 32) {                          // one wave issues the tensor ops
    tdm_issue_1d(src,      rowbuf,      b1);
    tdm_issue_1d(src + b1, rowbuf + b1, b2 - b1);
    tdm_issue_1d(src + b2, rowbuf + b2, b3 - b2);
    tdm_issue_1d(src + b3, rowbuf + b3, V - b3);
  }
  // Pipeline: histogram chunk k while chunks k+1.. are still in flight.
  if (tid < 32) __builtin_amdgcn_s_wait_tensorcnt(3);
  __syncthreads();
  pass1(0, b1);
  if (tid < 32) __builtin_amdgcn_s_wait_tensorcnt(2);
  __syncthreads();
  pass1(b1, b2);
  if (tid < 32) __builtin_amdgcn_s_wait_tensorcnt(1);
  __syncthreads();
  pass1(b2, b3);
  if (tid < 32) __builtin_amdgcn_s_wait_tensorcnt(0);
  __syncthreads();
  pass1(b3, V);
#else
  __syncthreads();
  for (int v = tid; v < V; v += BLOCK) rowbuf[v] = src[v];
  __syncthreads();
  pass1(0, V);
#endif

  atomicAdd(&s_sum, lsum);
  atomicMax(&s_maxbits, lmax);
  __syncthreads();

  const float budget = TOPP * s_sum;

  // ---- Coarse cut: boundary bin b* and residual mass R (parallel scan) ----
  find_cut(hist, budget, s_suf, &s_bstar, &s_R, tid);
  const int bstar = s_bstar;
  const float R = s_R;

  // ---- Refinement: histogram boundary-bin elements on bits[16:4] ----
  for (int b = tid; b < NBINS; b += BLOCK) hist[b] = 0.0f;
  __syncthreads();
  for (int v = tid; v < V; v += BLOCK) {
    float p = rowbuf[v];
    unsigned bits = __float_as_uint(p);
    if ((int)(bits >> 17) == bstar)
      atomicAdd(&hist[(bits >> 4) & (NBINS - 1)], p);
  }
  __syncthreads();
  find_cut(hist, R, s_suf, &s_bstar, &s_R, tid);
  const unsigned K = ((unsigned)bstar << 13) | (unsigned)s_bstar;  // 28-bit threshold
  const unsigned maxbits = s_maxbits;

  // ---- Final pass: Gumbel-max over the nucleus (log S shift cancels) ----
  float best = -1e30f; int bidx = 0;
  const unsigned long long rbase = (unsigned long long)rowId * (unsigned long long)V;
  for (int v = tid; v < V; v += BLOCK) {
    float p = rowbuf[v];
    unsigned bits = __float_as_uint(p);
    if ((bits >> 4) >= K || bits == maxbits) {   // always keep the top token
      float sc = __logf(p) + gumbel_at(rbase + (unsigned long long)v);
      if (sc > best) { best = sc; bidx = v; }
    }
  }
  for (int off = 16; off > 0; off >>= 1) {       // wave32 reduction
    float ob = __shfl_down(best, off, 32);
    int   oi = __shfl_down(bidx, off, 32);
    if (ob > best) { best = ob; bidx = oi; }
  }
  if ((tid & 31) == 0) { s_wbest[tid >> 5] = best; s_wbidx[tid >> 5] = bidx; }
  __syncthreads();
  if (tid == 0) {
    float bb = s_wbest[0]; int bi = s_wbidx[0];
    for (int w = 1; w < BLOCK / 32; ++w)
      if (s_wbest[w] > bb) { bb = s_wbest[w]; bi = s_wbidx[w]; }
    out[rowId] = bi;
  }
}

extern "C" void kernel_launch(void* const* d_in, const int* in_sizes, int n_in,
                              void* d_out, int out_size, void* d_ws, size_t ws_size,
                              hipStream_t stream) {
  (void)d_ws; (void)ws_size;
  const float* probs = (const float*)d_in[0];
  // d_in[1] (coherence) cancels analytically: w = p*(1+c)/((1+c)*sum p) = p/sum p,
  // and the -log(sum) shift is argmax-invariant under Gumbel-max.
  int rows = (n_in > 1 && in_sizes[1] > 0) ? in_sizes[1] : out_size;  // B*S
  if (rows <= 0) rows = 1;
  int V = in_sizes[0] / rows;                                         // 50257
  int* out = (int*)d_out;

  size_t shmem = ((size_t)V + NBINS) * sizeof(float);   // ~234 KB of the 320 KB WGP LDS
  (void)hipFuncSetAttribute((const void*)nucleus_sample_kernel,
                            hipFuncAttributeMaxDynamicSharedMemorySize, (int)shmem);
  hipLaunchKernelGGL(nucleus_sample_kernel, dim3(rows), dim3(BLOCK), shmem, stream,
                     probs, V, out);
}